// AudioVisualAlignmentLoss_41601053229348
// MI455X (gfx1250) — compile-verified
//
#include <hip/hip_runtime.h>
#include <hip/hip_bf16.h>
#include <math.h>

// Problem constants (from reference): B=32, T=8192, S=64, D=128, E=128
#define B_  32
#define T_  8192
#define S_  64
#define D_  128
#define E_  128
#define BS_ (B_ * S_)          // 2048 (b,s) rows

typedef float v2f __attribute__((ext_vector_type(2)));
typedef float v4f __attribute__((ext_vector_type(4)));
typedef float v8f __attribute__((ext_vector_type(8)));

// ---------------------------------------------------------------------------
// Kernel 1: masked segment mean-pool of audio. One block per (b,s).
// 8 waves per block; each wave loads one timestep's 128 floats as 32 x v4f
// (coalesced 512B per wave), striding by 8 timesteps. Nontemporal loads:
// audio is 128MB stream-once data.
// ---------------------------------------------------------------------------
__global__ __launch_bounds__(256) void seg_pool_kernel(
    const float* __restrict__ audio,   // (B,T,D)
    const int*   __restrict__ mask,    // (B,T)
    const int*   __restrict__ length,  // (B,)
    float*       __restrict__ audio_seg, // (B*S, D)
    float*       __restrict__ accum)     // [0]=num, [1]=den (zeroed here)
{
    const int bs = blockIdx.x;     // 0..2047
    const int b  = bs >> 6;
    const int s  = bs & 63;

    if (bs == 0 && threadIdx.x == 0) { accum[0] = 0.0f; accum[1] = 0.0f; }

    const int   L    = length[b];
    const float step = (float)L * (1.0f / 64.0f);
    // bounds = linspace(0, L, 65).astype(int32); endpoint is exactly L
    const int start = (s == 0)  ? 0 : (int)((float)s * step);
    const int end   = (s == 63) ? L : (int)((float)(s + 1) * step);

    const int lane = threadIdx.x & 31;
    const int wave = threadIdx.x >> 5;   // 0..7

    const int*   mrow  = mask + b * T_;
    const float* abase = audio + (size_t)b * T_ * D_;

    v4f sum = {0.f, 0.f, 0.f, 0.f};
    int cnt = 0;
    for (int t = start + wave; t < end; t += 8) {
        if (mrow[t]) {
            const v4f* p = (const v4f*)(abase + (size_t)t * D_) + lane;
            v4f v = __builtin_nontemporal_load(p);
            sum += v;
            cnt++;
        }
    }

    __shared__ v4f smem[8][32];
    __shared__ int scnt[8];
    smem[wave][lane] = sum;
    if (lane == 0) scnt[wave] = cnt;
    __syncthreads();

    if (wave == 0) {
        v4f tot = smem[0][lane];
        int c = scnt[0];
        #pragma unroll
        for (int w = 1; w < 8; ++w) {
            tot += smem[w][lane];
            c += scnt[w];
        }
        const float inv = (c > 0) ? (1.0f / (float)c) : 0.0f;
        tot *= inv;
        ((v4f*)(audio_seg + (size_t)bs * D_))[lane] = tot;
    }
}

// ---------------------------------------------------------------------------
// Kernel 2: x = audio_seg @ W^T + b_lin  via V_WMMA_F32_16X16X4_F32,
// then LayerNorm(E) -> L2 normalize -> cosine vs visual_embed -> weighted
// accumulation.  128 blocks x 8 waves; block g owns rows [16g, 16g+16);
// wave w owns output-channel tile [16w, 16w+16). K=128 -> 32 WMMA per wave.
//
// Fragment layouts (ISA 7.12.2):
//   A 16x4 f32:  lane l -> M = l&15 ; VGPR g holds K = g + 2*(l>>4)
//   B 4x16 f32:  lane l -> N = l&15 ; VGPR g holds K = g + 2*(l>>4)
//   C/D 16x16:   VGPR r: lanes 0-15 -> (M=r, N=lane), lanes 16-31 -> (M=r+8)
// ---------------------------------------------------------------------------
#define XS_STRIDE 132   // 128 + pad to avoid LDS bank conflicts

__global__ __launch_bounds__(256) void gemm_ln_cos_kernel(
    const float* __restrict__ audio_seg,  // (B*S, D)
    const float* __restrict__ W,          // (E, D)
    const float* __restrict__ b_lin,      // (E,)
    const float* __restrict__ gamma,      // (E,)
    const float* __restrict__ beta,       // (E,)
    const float* __restrict__ visual,     // (B*S, E)
    const float* __restrict__ sync_score, // (B*S,)
    const int*   __restrict__ seg_mask,   // (B*S,)
    float*       __restrict__ accum)      // [0]=num, [1]=den
{
    __shared__ float xs[16][XS_STRIDE];

    const int g    = blockIdx.x;        // 0..127
    const int tid  = threadIdx.x;
    const int wave = tid >> 5;          // 0..7 -> output channel tile
    const int lane = tid & 31;
    const int half = lane >> 4;         // 0/1
    const int sub  = lane & 15;
    const int r0   = g * 16;            // first bs row of this block
    const int n0   = wave * 16;         // first output channel of this wave

    // ---- GEMM phase: c[16x16] = A[16x128] * B[128x16] ----
    const float* Arow = audio_seg + (size_t)(r0 + sub) * D_ + half * 2;
    const float* Brow = W         + (size_t)(n0 + sub) * D_ + half * 2;

    v8f c = {0.f, 0.f, 0.f, 0.f, 0.f, 0.f, 0.f, 0.f};
    #pragma unroll
    for (int k0 = 0; k0 < D_; k0 += 4) {
        v2f a, bb;
        a[0]  = Arow[k0];  a[1]  = Arow[k0 + 1];
        bb[0] = Brow[k0];  bb[1] = Brow[k0 + 1];
        c = __builtin_amdgcn_wmma_f32_16x16x4_f32(
                /*neg_a=*/false, a, /*neg_b=*/false, bb,
                /*c_mod=*/(short)0, c, /*reuse_a=*/false, /*reuse_b=*/false);
    }

    // Scatter this wave's 16x16 tile into the 16x128 block tile in LDS.
    #pragma unroll
    for (int r = 0; r < 8; ++r) {
        xs[r + 8 * half][n0 + sub] = c[r];
    }
    __syncthreads();

    // ---- LN + normalize + cosine phase ----
    // Each half-wave (16 lanes) owns one row; lane `sub` owns 8 channels.
    const int row = 2 * wave + half;    // 0..15
    const int bs  = r0 + row;

    float y[8];
    float s1 = 0.f, s2 = 0.f;
    #pragma unroll
    for (int j = 0; j < 8; ++j) {
        const int n = sub * 8 + j;
        const float v = xs[row][n] + b_lin[n];
        y[j] = v;
        s1 += v;
        s2 += v * v;
    }
    #pragma unroll
    for (int m = 1; m < 16; m <<= 1) {
        s1 += __shfl_xor(s1, m, 16);
        s2 += __shfl_xor(s2, m, 16);
    }
    const float mu   = s1 * (1.0f / 128.0f);
    const float var  = s2 * (1.0f / 128.0f) - mu * mu;
    const float rstd = rsqrtf(var + 1e-5f);

    float n2 = 0.f;
    #pragma unroll
    for (int j = 0; j < 8; ++j) {
        const int n = sub * 8 + j;
        y[j] = (y[j] - mu) * rstd * gamma[n] + beta[n];
        n2 += y[j] * y[j];
    }
    #pragma unroll
    for (int m = 1; m < 16; m <<= 1) n2 += __shfl_xor(n2, m, 16);

    const float inv = 1.0f / fmaxf(sqrtf(n2), 1e-12f);

    float cp = 0.f;
    const float* ve = visual + (size_t)bs * E_;
    #pragma unroll
    for (int j = 0; j < 8; ++j) {
        cp += y[j] * ve[sub * 8 + j];
    }
    #pragma unroll
    for (int m = 1; m < 16; m <<= 1) cp += __shfl_xor(cp, m, 16);

    if (sub == 0) {
        const float weight = sync_score[bs] * (float)seg_mask[bs];
        atomicAdd(&accum[0], cp * inv * weight);
        atomicAdd(&accum[1], weight);
    }
}

// ---------------------------------------------------------------------------
// Kernel 3: finalize loss = -num / max(den, 1e-6)
// ---------------------------------------------------------------------------
__global__ void finalize_kernel(const float* __restrict__ accum,
                                float* __restrict__ out)
{
    out[0] = -accum[0] / fmaxf(accum[1], 1e-6f);
}

// ---------------------------------------------------------------------------
extern "C" void kernel_launch(void* const* d_in, const int* in_sizes, int n_in,
                              void* d_out, int out_size, void* d_ws, size_t ws_size,
                              hipStream_t stream)
{
    const float* audio      = (const float*)d_in[0];  // (B,T,D)
    const float* visual     = (const float*)d_in[1];  // (B,S,E)
    const float* sync_score = (const float*)d_in[2];  // (B,S)
    const int*   mask       = (const int*)  d_in[3];  // (B,T)
    const int*   length     = (const int*)  d_in[4];  // (B,)
    const int*   seg_mask   = (const int*)  d_in[5];  // (B,S)
    const float* W          = (const float*)d_in[6];  // (E,D)
    const float* b_lin      = (const float*)d_in[7];  // (E,)
    const float* gamma      = (const float*)d_in[8];  // (E,)
    const float* beta       = (const float*)d_in[9];  // (E,)

    float* ws        = (float*)d_ws;
    float* accum     = ws;          // 2 floats (+ pad)
    float* audio_seg = ws + 64;     // 256B-aligned, B*S*D floats = 1MB

    seg_pool_kernel<<<BS_, 256, 0, stream>>>(audio, mask, length, audio_seg, accum);
    gemm_ln_cos_kernel<<<BS_ / 16, 256, 0, stream>>>(audio_seg, W, b_lin, gamma, beta,
                                                     visual, sync_score, seg_mask, accum);
    finalize_kernel<<<1, 1, 0, stream>>>(accum, (float*)d_out);
}